// Embedding_22814866276768
// MI455X (gfx1250) — compile-verified
//
#include <hip/hip_runtime.h>
#include <hip/hip_bf16.h>

// ---------------------------------------------------------------------------
// Embedding gather via CDNA5 Tensor Data Mover (TDM) in gather mode.
//   out[t, :] = weight[x[t], :]   t in [0, 8192), row = 512 f32 = 2 KB
//
// One TDM descriptor (32-bit index mode) gathers 8 indexed rows (16 KB)
// from the table into LDS; a second descriptor streams the 16 KB tile
// (8 consecutive output rows are contiguous) back to global memory.
// TENSORcnt retires in order per wave, so with 3 LDS buffers we can keep
// the next gather in flight across each store (wait tensorcnt<=1).
// Index fetch: lanes 0..7 load the 8 ids in ONE global_load_b32, then
// v_readlane lifts them into SGPRs for the descriptor.
// ---------------------------------------------------------------------------

typedef __attribute__((ext_vector_type(4))) unsigned int u32x4;
typedef __attribute__((ext_vector_type(8))) int          i32x8;
typedef __attribute__((ext_vector_type(4))) int          i32x4;

#define EMB_ROW_ELEMS   512          // floats per embedding row
#define EMB_NUM_ROWS    32000        // rows in the table (OOB bound)
#define ROWS_PER_CHUNK  8            // rows per TDM gather descriptor (32b idx mode max)
#define ROW_BYTES       (EMB_ROW_ELEMS * 4)
#define CHUNK_BYTES     (ROWS_PER_CHUNK * ROW_BYTES)   // 16 KiB
#define NBUF            3            // LDS buffers per wave (pipeline depth 2)
#define WAVES_PER_WG    2
#define WG_THREADS      (WAVES_PER_WG * 32)            // wave32

// --- TDM builtin wrappers (arity differs between toolchains, see bridge doc)
static __device__ __forceinline__
void tdm_load(u32x4 g0, i32x8 g1, i32x4 g2, i32x4 g3) {
#if defined(__clang_major__) && (__clang_major__ >= 23)
    i32x8 z = 0;
    __builtin_amdgcn_tensor_load_to_lds(g0, g1, g2, g3, z, 0);
#else
    __builtin_amdgcn_tensor_load_to_lds(g0, g1, g2, g3, 0);
#endif
}

static __device__ __forceinline__
void tdm_store(u32x4 g0, i32x8 g1, i32x4 g2, i32x4 g3) {
#if defined(__clang_major__) && (__clang_major__ >= 23)
    i32x8 z = 0;
    __builtin_amdgcn_tensor_store_from_lds(g0, g1, g2, g3, z, 0);
#else
    __builtin_amdgcn_tensor_store_from_lds(g0, g1, g2, g3, 0);
#endif
}

// --- Gather-mode load: pull `nrows` indexed 512-f32 rows into LDS starting at byte offset lds_off
static __device__ __forceinline__
void tdm_gather_rows_to_lds(unsigned lds_off, const float* table,
                            const int idx[8], int nrows) {
    unsigned long long ga = (unsigned long long)(const void*)table;
    // D# group0: count=1 | gather_index_size=32b (bit30) | gather_mode (bit31)
    u32x4 g0 = { 0xC0000001u,
                 lds_off,
                 (unsigned)(ga & 0xFFFFFFFFull),
                 (unsigned)((ga >> 32) & 0x01FFFFFFull) | 0x80000000u /*type=2*/ };
    // D# group1
    i32x8 g1 = {
        0x00020000,                                                 // data_size=4B
        (int)(((unsigned)EMB_ROW_ELEMS & 0xFFFFu) << 16),           // tensor_dim0 lo16
        (int)((((unsigned)EMB_ROW_ELEMS >> 16) & 0xFFFFu) |
              (((unsigned)EMB_NUM_ROWS & 0xFFFFu) << 16)),          // dim0 hi | dim1 lo
        (int)((((unsigned)EMB_NUM_ROWS >> 16) & 0xFFFFu) |
              (((unsigned)EMB_ROW_ELEMS & 0xFFFFu) << 16)),         // dim1 hi | tile_dim0
        (int)(nrows & 0xFFFF),                                      // tile_dim1 = #indices
        (int)EMB_ROW_ELEMS,                                         // tensor_dim0_stride lo32
        0, 0 };
    i32x4 g2 = { idx[0], idx[1], idx[2], idx[3] };                  // row_index_0..3
    i32x4 g3 = { idx[4], idx[5], idx[6], idx[7] };                  // row_index_4..7
    tdm_load(g0, g1, g2, g3);
}

// --- Normal-mode store: copy `elems` contiguous f32 from LDS at byte offset lds_off to dst
static __device__ __forceinline__
void tdm_store_lds_to_global(unsigned lds_off, float* dst, int elems) {
    unsigned long long ga = (unsigned long long)(void*)dst;
    u32x4 g0 = { 0x00000001u,                                       // count=1
                 lds_off,
                 (unsigned)(ga & 0xFFFFFFFFull),
                 (unsigned)((ga >> 32) & 0x01FFFFFFull) | 0x80000000u /*type=2*/ };
    i32x8 g1 = {
        0x00020000,                                                 // data_size=4B
        (int)(((unsigned)elems & 0xFFFFu) << 16),                   // tensor_dim0 lo16
        (int)((((unsigned)elems >> 16) & 0xFFFFu) | (1u << 16)),    // dim0 hi | dim1=1
        (int)(((unsigned)elems & 0xFFFFu) << 16),                   // tile_dim0 = elems
        1,                                                          // tile_dim1 = 1
        elems,                                                      // tensor_dim0_stride
        0, 0 };
    i32x4 z = { 0, 0, 0, 0 };
    tdm_store(g0, g1, z, z);
}

__global__ void Embedding_22814866276768_kernel(const int* __restrict__ x,
                                                const float* __restrict__ w,
                                                float* __restrict__ out,
                                                int ntok) {
    extern __shared__ char lds_buf[];                 // WAVES_PER_WG * NBUF * 16 KiB
    (void)lds_buf;

    const int lane       = (int)(threadIdx.x & 31);
    const int wave       = (int)(threadIdx.x >> 5);
    const int gwave      = (int)blockIdx.x * WAVES_PER_WG + wave;
    const int nwaves     = (int)gridDim.x * WAVES_PER_WG;
    const int nchunks    = (ntok + ROWS_PER_CHUNK - 1) / ROWS_PER_CHUNK;
    const unsigned lbase = (unsigned)wave * ((unsigned)NBUF * CHUNK_BYTES);

    // Issue a gather for chunk c into LDS offset `off`; returns rows gathered.
    // All 32 lanes load one id each (duplicated/clamped) -> ONE global_load_b32,
    // then v_readlane pulls ids 0..7 into SGPRs for the descriptor.
    auto issue_gather = [&](int c, unsigned off) -> int {
        const int base  = c * ROWS_PER_CHUNK;
        int nrows = ntok - base;
        if (nrows > ROWS_PER_CHUNK) nrows = ROWS_PER_CHUNK;
        int k = lane & (ROWS_PER_CHUNK - 1);
        if (k >= nrows) k = nrows - 1;                // branchless tail clamp
        const int v = x[base + k];                    // coalesced 32B load
        int idx[8];
#pragma unroll
        for (int i = 0; i < 8; ++i)
            idx[i] = __builtin_amdgcn_readlane(v, i); // -> SGPRs
        tdm_gather_rows_to_lds(off, w, idx, nrows);
        return nrows;
    };

    int c = gwave;
    if (c >= nchunks) return;

    int bk = 0;                                       // buffer of chunk c
    int nr = issue_gather(c, lbase);                  // G(c) in flight
    for (;;) {
        const int  cn        = c + nwaves;
        const bool have_next = (cn < nchunks);
        const int  bn        = (bk == NBUF - 1) ? 0 : bk + 1;
        int nr_next = 0;
        if (have_next)                                // G(cn) BEFORE the wait:
            nr_next = issue_gather(cn, lbase + (unsigned)bn * CHUNK_BYTES);
        if (have_next)
            __builtin_amdgcn_s_wait_tensorcnt(1);     // in-order: G(c)+older stores done
        else
            __builtin_amdgcn_s_wait_tensorcnt(0);
        tdm_store_lds_to_global(lbase + (unsigned)bk * CHUNK_BYTES,
                                out + (size_t)c * (ROWS_PER_CHUNK * EMB_ROW_ELEMS),
                                nr * EMB_ROW_ELEMS);
        if (!have_next) break;
        bk = bn;
        c  = cn;
        nr = nr_next;
    }
    __builtin_amdgcn_s_wait_tensorcnt(0);             // all stores retired
}

extern "C" void kernel_launch(void* const* d_in, const int* in_sizes, int n_in,
                              void* d_out, int out_size, void* d_ws, size_t ws_size,
                              hipStream_t stream) {
    (void)n_in; (void)out_size; (void)d_ws; (void)ws_size;
    const int*   x   = (const int*)d_in[0];     // [4, 2048] int32 token ids
    const float* w   = (const float*)d_in[1];   // [32000, 512] f32 table
    float*       out = (float*)d_out;           // [4, 2048, 512] f32

    const int ntok    = in_sizes[0];                              // 8192
    const int nchunks = (ntok + ROWS_PER_CHUNK - 1) / ROWS_PER_CHUNK;
    // ~2 chunks per wave stream -> 512 concurrent TDM pipelines at default size
    int blocks = (nchunks + 2 * WAVES_PER_WG - 1) / (2 * WAVES_PER_WG);
    if (blocks < 1)    blocks = 1;
    if (blocks > 2048) blocks = 2048;
    const size_t shmem = (size_t)WAVES_PER_WG * NBUF * CHUNK_BYTES; // 96 KiB

    Embedding_22814866276768_kernel<<<blocks, WG_THREADS, shmem, stream>>>(x, w, out, ntok);
}